// Adaptive_temperature_scale_82205674045819
// MI455X (gfx1250) — compile-verified
//
#include <hip/hip_runtime.h>
#include <math.h>

// ---------------------------------------------------------------------------
// Adaptive temperature scaling loss, MI455X (gfx1250, wave32).
// HBM-bound: 262 MB single pass. Rows staged to LDS via async global->LDS
// DMA (ASYNCcnt), 3 logical passes run out of LDS, wave32 shuffle reductions.
// ---------------------------------------------------------------------------

#define THREADS        256
#define ROWS_PER_BLOCK 8            // one wave32 per row
#define C_DIM          1000
#define CHUNKS         250          // C_DIM / 4 (float4 chunks per row)
#define LANE_ITERS     8            // ceil(CHUNKS / 32)
#define LOGC_INV       0.14476482730108395f   // 1 / ln(1000)
#define EPS_F          1.1920928955078125e-07f
#define NEG_HUGE       -3.402823466e+38f

#if defined(__has_builtin)
#if __has_builtin(__builtin_amdgcn_global_load_async_to_lds_b128) && \
    __has_builtin(__builtin_amdgcn_s_wait_asynccnt)
#define HAVE_ASYNC_LDS 1
#endif
#endif
#ifndef HAVE_ASYNC_LDS
#define HAVE_ASYNC_LDS 0
#endif

#if HAVE_ASYNC_LDS
// Match the builtin's expected parameter types exactly:
//   param0: 'vector_size(16) int __device__ *'  (addrspace(1))
//   param1: LDS-side pointer (addrspace(3))
typedef int vec4i __attribute__((vector_size(16)));
typedef vec4i __attribute__((address_space(1))) g_vec4i;
typedef vec4i __attribute__((address_space(3))) l_vec4i;
#endif

__device__ __forceinline__ float wave_max(float v) {
#pragma unroll
  for (int o = 16; o > 0; o >>= 1) v = fmaxf(v, __shfl_xor(v, o, 32));
  return v;
}
__device__ __forceinline__ float wave_sum(float v) {
#pragma unroll
  for (int o = 16; o > 0; o >>= 1) v += __shfl_xor(v, o, 32);
  return v;
}

// Map a staging-chunk id to (global src, LDS dst). Chunks 0..8*250-1 are the
// 8 x-rows; chunks 8*250..8*250+249 are the shared w_L copy.
__device__ __forceinline__ void stage_addrs(int c, int row0,
                                            const float* __restrict__ X,
                                            const float* __restrict__ wL,
                                            float* smem,
                                            const float** g, float** l) {
  if (c < ROWS_PER_BLOCK * CHUNKS) {
    int row = c / CHUNKS;
    int ch  = c - row * CHUNKS;
    *g = X + (size_t)(row0 + row) * C_DIM + ch * 4;
    *l = smem + row * C_DIM + ch * 4;
  } else {
    int ch = c - ROWS_PER_BLOCK * CHUNKS;
    *g = wL + ch * 4;
    *l = smem + ROWS_PER_BLOCK * C_DIM + ch * 4;
  }
}

__global__ __launch_bounds__(THREADS) void atts_rows_kernel(
    const float* __restrict__ X, const int* __restrict__ labels,
    const float* __restrict__ wL, const float* __restrict__ wH,
    const float* __restrict__ bias, float* __restrict__ partial) {
  // 8 rows of x (32 KB) + one copy of w_L (4 KB)
  __shared__ float smem[ROWS_PER_BLOCK * C_DIM + C_DIM];
  __shared__ float row_nll[ROWS_PER_BLOCK];

  const int tid  = threadIdx.x;
  const int row0 = blockIdx.x * ROWS_PER_BLOCK;
  const int total_chunks = ROWS_PER_BLOCK * CHUNKS + CHUNKS;

  // ---- Stage block's rows + w_L into LDS (async DMA path on gfx1250) ----
#if HAVE_ASYNC_LDS
  for (int c = tid; c < total_chunks; c += THREADS) {
    const float* g; float* l;
    stage_addrs(c, row0, X, wL, smem, &g, &l);
    __builtin_amdgcn_global_load_async_to_lds_b128(
        (g_vec4i*)g, (l_vec4i*)l, /*offset=*/0, /*cpol=*/0);
  }
  __builtin_amdgcn_s_wait_asynccnt(0);
#else
  for (int c = tid; c < total_chunks; c += THREADS) {
    const float* g; float* l;
    stage_addrs(c, row0, X, wL, smem, &g, &l);
    *(float4*)l = *(const float4*)g;
  }
#endif
  __syncthreads();

  const int wave = tid >> 5;
  const int lane = tid & 31;
  const float* xr = smem + wave * C_DIM;                 // this wave's row
  const float* wr = smem + ROWS_PER_BLOCK * C_DIM;       // shared w_L
  const int grow  = row0 + wave;

  // ---- Pass A: row max + LTS = x . w_L ----
  float m = NEG_HUGE, lts = 0.f;
#pragma unroll
  for (int j = 0; j < LANE_ITERS; ++j) {
    int idx = lane + 32 * j;
    if (idx < CHUNKS) {
      float4 x4 = *(const float4*)(xr + 4 * idx);
      float4 w4 = *(const float4*)(wr + 4 * idx);
      lts = fmaf(x4.x, w4.x, lts);
      lts = fmaf(x4.y, w4.y, lts);
      lts = fmaf(x4.z, w4.z, lts);
      lts = fmaf(x4.w, w4.w, lts);
      m = fmaxf(m, fmaxf(fmaxf(x4.x, x4.y), fmaxf(x4.z, x4.w)));
    }
  }
  m   = wave_max(m);
  lts = wave_sum(lts);

  // ---- Pass B: S = sum e^(x-m), Tn = sum (x-m) e^(x-m) ----
  float s = 0.f, tn = 0.f;
#pragma unroll
  for (int j = 0; j < LANE_ITERS; ++j) {
    int idx = lane + 32 * j;
    if (idx < CHUNKS) {
      float4 x4 = *(const float4*)(xr + 4 * idx);
      float d, e;
      d = x4.x - m; e = __expf(d); s += e; tn = fmaf(d, e, tn);
      d = x4.y - m; e = __expf(d); s += e; tn = fmaf(d, e, tn);
      d = x4.z - m; e = __expf(d); s += e; tn = fmaf(d, e, tn);
      d = x4.w - m; e = __expf(d); s += e; tn = fmaf(d, e, tn);
    }
  }
  s  = wave_sum(s);
  tn = wave_sum(tn);

  // H_hat = sum p log p = Tn/S - log S   (negative entropy)
  float H   = tn / s - __logf(s);
  float hts = wH[0] * (H * LOGC_INV) + bias[0];
  float a   = lts + hts;
  // stable softplus, clipped to EPS
  float sp   = fmaxf(a, 0.f) + log1pf(__expf(-fabsf(a)));
  float T    = fmaxf(sp, EPS_F);
  float invT = 1.0f / T;

  // ---- Pass C: S2 = sum e^((x-m)/T) ----
  float s2 = 0.f;
#pragma unroll
  for (int j = 0; j < LANE_ITERS; ++j) {
    int idx = lane + 32 * j;
    if (idx < CHUNKS) {
      float4 x4 = *(const float4*)(xr + 4 * idx);
      s2 += __expf((x4.x - m) * invT);
      s2 += __expf((x4.y - m) * invT);
      s2 += __expf((x4.z - m) * invT);
      s2 += __expf((x4.w - m) * invT);
    }
  }
  s2 = wave_sum(s2);

  // nll = log S2 - (x[label]-m)/T ; row is LDS-resident so lookup is free
  int   lab = labels[grow];
  float nll = __logf(s2) - (xr[lab] - m) * invT;

  if (lane == 0) row_nll[wave] = nll;
  __syncthreads();
  if (tid == 0) {
    float acc = 0.f;
#pragma unroll
    for (int r = 0; r < ROWS_PER_BLOCK; ++r) acc += row_nll[r];
    partial[blockIdx.x] = acc;
  }
}

// Deterministic final reduction: fixed single block, fixed tree shape.
__global__ __launch_bounds__(256) void atts_reduce_kernel(
    const float* __restrict__ partial, int n, float inv_total,
    float* __restrict__ out) {
  __shared__ float red[256];
  float acc = 0.f;
  for (int i = threadIdx.x; i < n; i += 256) acc += partial[i];
  red[threadIdx.x] = acc;
  __syncthreads();
#pragma unroll
  for (int s = 128; s > 0; s >>= 1) {
    if (threadIdx.x < s) red[threadIdx.x] += red[threadIdx.x + s];
    __syncthreads();
  }
  if (threadIdx.x == 0) out[0] = red[0] * inv_total;
}

extern "C" void kernel_launch(void* const* d_in, const int* in_sizes, int n_in,
                              void* d_out, int out_size, void* d_ws, size_t ws_size,
                              hipStream_t stream) {
  const float* X      = (const float*)d_in[0];
  const int*   labels = (const int*)d_in[1];
  const float* wL     = (const float*)d_in[2];
  const float* wH     = (const float*)d_in[3];
  const float* bias   = (const float*)d_in[4];

  const int Brows   = in_sizes[1];                 // 65536 rows
  const int nblocks = Brows / ROWS_PER_BLOCK;      // 8192 partials (32 KB ws)

  float* partial = (float*)d_ws;

  atts_rows_kernel<<<nblocks, THREADS, 0, stream>>>(X, labels, wL, wH, bias,
                                                    partial);
  atts_reduce_kernel<<<1, 256, 0, stream>>>(partial, nblocks,
                                            1.0f / (float)Brows,
                                            (float*)d_out);
}